// PatchProcessor_4767413698961
// MI455X (gfx1250) — compile-verified
//
#include <hip/hip_runtime.h>
#include <cstdint>

// out[b,c,h,w] = x[b,c,h,w] * cnt/(cnt+1e-8), cnt = cr(h)*cc(w),
// cr(h)=2 iff 32<=h<480 else 1 (same for cc). Pure streaming scale:
// ~50 MB HBM traffic -> ~2.2 us floor at 23.3 TB/s. We stream through the
// CDNA5 async global->LDS path (ASYNCcnt) with a 2-stage double buffer.

typedef float f32x4 __attribute__((ext_vector_type(4)));

#define TPB    256
#define ITERS  4
#define CPB    (TPB * ITERS)   // float4 chunks per block = 1024

// Hard compiler-level ordering: asm + "memory" clobber guarantees the
// ds_load of the staging buffer cannot be hoisted above the asynccnt drain
// (a wait *builtin* is side-effecting but not necessarily memory-writing,
// so LLVM could in principle reorder LDS loads across it).
#define WAIT_ASYNC_LE1() asm volatile("s_wait_asynccnt 0x1" ::: "memory")
#define WAIT_ASYNC_0()   asm volatile("s_wait_asynccnt 0x0" ::: "memory")

// Issue one 128-bit async DMA from global memory into LDS (per lane).
// LDS byte offset = low 32 bits of the flat shared-aperture address
// (ISA: generic->LDS mapping truncates the upper bits; verified to
// const-fold to the plain workgroup-relative offset in the emitted asm).
__device__ __forceinline__ void async_load_b128(void* lds_dst, const f32x4* gsrc) {
  unsigned lds_off = (unsigned)(uintptr_t)lds_dst;
  asm volatile("global_load_async_to_lds_b128 %0, %1, off"
               :
               : "v"(lds_off), "v"(gsrc)
               : "memory");
}

__global__ __launch_bounds__(TPB)
void patch_overlap_norm(const f32x4* __restrict__ in, f32x4* __restrict__ out) {
  __shared__ f32x4 buf[2][TPB];
  const int tid  = threadIdx.x;
  const int base = (int)blockIdx.x * CPB + tid;

  // Prologue: stage 0 in flight.
  async_load_b128(&buf[0][tid], in + base);

#pragma unroll
  for (int it = 0; it < ITERS; ++it) {
    if (it + 1 < ITERS) {
      // Keep the pipe full: issue stage it+1, then wait for stage it only
      // (async loads complete in order; <=1 outstanding => oldest is done).
      async_load_b128(&buf[(it + 1) & 1][tid], in + base + (it + 1) * TPB);
      WAIT_ASYNC_LE1();
    } else {
      WAIT_ASYNC_0();
    }

    f32x4 v = buf[it & 1][tid];   // ds_load_b128 (thread-private slot, no barrier)

    const int k  = base + it * TPB;   // float4-chunk index
    const int h  = (k >> 7) & 511;    // row within 512x512 plane (128 chunks/row)
    const int w4 = k & 127;           // float4 column; w = 4*w4 (never straddles
                                      // a coverage boundary: multiples of 32)
    const int cr = ((unsigned)(h  - 32) < 448u) ? 2 : 1;   // rows  [32,480)
    const int cc = ((unsigned)(w4 -  8) < 112u) ? 2 : 1;   // cols  [32,480)
    const float c = (float)(cr * cc);
    const float s = c / (c + 1e-8f);  // denominator rounds to c in fp32; exact div
    v *= s;
    out[k] = v;                       // global_store_b128
  }
}

extern "C" void kernel_launch(void* const* d_in, const int* in_sizes, int n_in,
                              void* d_out, int out_size, void* d_ws, size_t ws_size,
                              hipStream_t stream) {
  (void)n_in; (void)out_size; (void)d_ws; (void)ws_size;
  const f32x4* in  = (const f32x4*)d_in[0];
  f32x4*       out = (f32x4*)d_out;
  const int chunks = in_sizes[0] / 4;   // 6,291,456 / 4 = 1,572,864
  const int blocks = chunks / CPB;      // 1,536 (exact)
  patch_overlap_norm<<<blocks, TPB, 0, stream>>>(in, out);
}